// LightGCNConv_49675591745620
// MI455X (gfx1250) — compile-verified
//
#include <hip/hip_runtime.h>
#include <cstdint>

// ---------------------------------------------------------------------------
// GraphConv(norm='both'): out = D_in^{-1/2} * A * (D_out^{-1/2} * emb)
// N = 100000, D = 64, E = 1.25M. L2-resident (emb 25.6MB + out 25.6MB << 192MB).
//
// CSR-gather formulation (avoids 80M f32 scatter atomics):
//   p1: zero degree counters (2N ints)
//   p2: out/in-degree histograms (2.5M int atomics)
//   p3: src_norm = rsqrt(clamp(out_deg,1))
//   p4: exclusive scan of in-degree -> rowptr (3 small kernels)
//   p5: cursor fill -> col[] = sources grouped by destination (1.25M int atomics)
//   p6: one wave32 per dst node: async global->LDS double-buffered row gather
//       (ASYNCcnt pipeline), FMA-accumulate in registers, single store of
//       out[n,:] * dst_norm[n]. Zero f32 atomics, zero output pre-pass.
// ---------------------------------------------------------------------------

#define FEAT 64

#if __has_builtin(__builtin_amdgcn_global_load_async_to_lds_b64) && \
    __has_builtin(__builtin_amdgcn_s_wait_asynccnt)
#define USE_ASYNC_LDS 1
#else
#define USE_ASYNC_LDS 0
#endif

#if USE_ASYNC_LDS
// b64 async op moves a <2 x i32> per lane; builtin takes AS1 src / AS3 dst
// pointers to that vector type (confirmed by compiler diagnostic in round 1).
typedef int v2i __attribute__((vector_size(8)));
typedef __attribute__((address_space(1))) v2i gv2i;
typedef __attribute__((address_space(3))) v2i lv2i;

__device__ __forceinline__ void async_gather8(const float* gptr, float* lptr) {
  // Drain our prior LDS reads first: closes the WAR window between the
  // previous iteration's ds_load of this buffer and the async write below.
#if __has_builtin(__builtin_amdgcn_s_wait_dscnt)
  __builtin_amdgcn_s_wait_dscnt(0);
#else
  asm volatile("s_wait_dscnt 0x0" ::: "memory");
#endif
  // 8 bytes/lane: wave32 moves one 256B emb row in one async instruction.
  __builtin_amdgcn_global_load_async_to_lds_b64(
      (gv2i*)(uintptr_t)gptr,
      (lv2i*)(unsigned int)(uintptr_t)lptr,
      /*offset=*/0, /*cpol=*/0);
}
#endif

// ---- p1: zero the two degree counter arrays -------------------------------
__global__ void k_zero_counts(int* __restrict__ cnts, int n) {
  int i = blockIdx.x * blockDim.x + threadIdx.x;
  if (i < n) cnts[i] = 0;
}

// ---- p2: degree histograms ------------------------------------------------
__global__ void k_count(const int* __restrict__ src, const int* __restrict__ dst,
                        int* __restrict__ out_cnt, int* __restrict__ in_cnt,
                        int E) {
  int i = blockIdx.x * blockDim.x + threadIdx.x;
  int stride = gridDim.x * blockDim.x;
  for (int e = i; e < E; e += stride) {
    atomicAdd(&out_cnt[src[e]], 1);   // global_atomic_add_u32
    atomicAdd(&in_cnt[dst[e]], 1);
  }
}

// ---- p3: src_norm = clamp(out_deg,1)^-1/2 ---------------------------------
__global__ void k_srcnorm(const int* __restrict__ out_cnt,
                          float* __restrict__ src_norm, int N) {
  int i = blockIdx.x * blockDim.x + threadIdx.x;
  if (i < N) src_norm[i] = rsqrtf(fmaxf((float)out_cnt[i], 1.0f));
}

// ---- p4: exclusive scan of in_cnt -> rowptr -------------------------------
#define SCAN_T 256
#define SCAN_V 4          // 1024 elements per block
__global__ void k_scan1(const int* __restrict__ cnt, int* __restrict__ pre,
                        int* __restrict__ bsum, int N) {
  __shared__ int sh[SCAN_T];
  const int base = blockIdx.x * (SCAN_T * SCAN_V) + threadIdx.x * SCAN_V;
  int v[SCAN_V];
  int s = 0;
  for (int k = 0; k < SCAN_V; ++k) {
    int idx = base + k;
    v[k] = (idx < N) ? cnt[idx] : 0;
    s += v[k];
  }
  sh[threadIdx.x] = s;
  __syncthreads();
  for (int off = 1; off < SCAN_T; off <<= 1) {   // Hillis-Steele inclusive
    int t = (threadIdx.x >= off) ? sh[threadIdx.x - off] : 0;
    __syncthreads();
    sh[threadIdx.x] += t;
    __syncthreads();
  }
  int run = sh[threadIdx.x] - s;                 // exclusive prefix in block
  if (threadIdx.x == SCAN_T - 1) bsum[blockIdx.x] = sh[SCAN_T - 1];
  for (int k = 0; k < SCAN_V; ++k) {
    int idx = base + k;
    if (idx < N) pre[idx] = run;
    run += v[k];
  }
}

__global__ void k_scan_top(int* __restrict__ bsum, int nb) {
  if (blockIdx.x == 0 && threadIdx.x == 0) {     // nb ~ 98: trivial serial scan
    int acc = 0;
    for (int i = 0; i < nb; ++i) { int t = bsum[i]; bsum[i] = acc; acc += t; }
  }
}

__global__ void k_scan2(int* __restrict__ rowptr, const int* __restrict__ bsum,
                        int* __restrict__ cursor, int N, int E) {
  int i = blockIdx.x * blockDim.x + threadIdx.x;
  if (i < N) {
    int r = rowptr[i] + bsum[i >> 10];           // 1024 elems per scan block
    rowptr[i] = r;
    cursor[i] = r;                               // insertion cursor = rowptr copy
    if (i == N - 1) rowptr[N] = E;
  }
}

// ---- p5: bucket-fill column indices grouped by destination ----------------
__global__ void k_fill(const int* __restrict__ src, const int* __restrict__ dst,
                       int* __restrict__ cursor, int* __restrict__ col, int E) {
  int i = blockIdx.x * blockDim.x + threadIdx.x;
  int stride = gridDim.x * blockDim.x;
  for (int e = i; e < E; e += stride) {
    int pos = atomicAdd(&cursor[dst[e]], 1);
    col[pos] = src[e];
  }
}

// ---- p6: per-node gather-accumulate (no f32 atomics) ----------------------
// One wave32 per destination node; lane d covers features [2d, 2d+1].
// Async double-buffered LDS staging hides random-row gather latency behind
// the FMA accumulation of the previous edge.
__global__ void k_gather(const float* __restrict__ emb,
                         const int* __restrict__ col,
                         const int* __restrict__ rowptr,
                         const float* __restrict__ src_norm,
                         float* __restrict__ out, int N) {
  const int lane = threadIdx.x & 31;
  const int wblk = threadIdx.x >> 5;
  const int wid  = blockIdx.x * (blockDim.x >> 5) + wblk;
  const int W    = gridDim.x * (blockDim.x >> 5);

#if USE_ASYNC_LDS
  __shared__ float lds[(256 / 32) * 2 * FEAT];   // 8 waves x 2 bufs x 64 f32
  float* buf = &lds[wblk * 2 * FEAT];
#endif

  for (int n = wid; n < N; n += W) {
    const int jb = rowptr[n];
    const int je = rowptr[n + 1];
    float ax = 0.0f, ay = 0.0f;

#if USE_ASYNC_LDS
    int cur = 0;
    if (jb < je) {
      int s0 = col[jb];
      async_gather8(emb + (long long)s0 * FEAT + lane * 2, buf + lane * 2);
    }
    for (int j = jb; j < je; ++j) {
      const int jn = j + 1;
      if (jn < je) {
        int sn = col[jn];
        async_gather8(emb + (long long)sn * FEAT + lane * 2,
                      buf + (cur ^ 1) * FEAT + lane * 2);
        __builtin_amdgcn_s_wait_asynccnt(1);     // current buffer landed
      } else {
        __builtin_amdgcn_s_wait_asynccnt(0);
      }
      const float nrm = src_norm[col[j]];
      float2 v = *(const float2*)(buf + cur * FEAT + lane * 2);  // ds_load_b64
      ax = fmaf(v.x, nrm, ax);
      ay = fmaf(v.y, nrm, ay);
      cur ^= 1;
    }
#else
    for (int j = jb; j < je; ++j) {
      const int s = col[j];
      const float nrm = src_norm[s];
      float2 v = *(const float2*)(emb + (long long)s * FEAT + lane * 2);
      ax = fmaf(v.x, nrm, ax);
      ay = fmaf(v.y, nrm, ay);
    }
#endif

    const float dn = rsqrtf(fmaxf((float)(je - jb), 1.0f));  // dst_norm
    float2 o;
    o.x = ax * dn;
    o.y = ay * dn;
    *(float2*)(out + (long long)n * FEAT + lane * 2) = o;
  }
}

extern "C" void kernel_launch(void* const* d_in, const int* in_sizes, int n_in,
                              void* d_out, int out_size, void* d_ws, size_t ws_size,
                              hipStream_t stream) {
  const float* emb = (const float*)d_in[0];
  const int*   src = (const int*)d_in[1];
  const int*   dst = (const int*)d_in[2];
  float*       out = (float*)d_out;

  const int N = in_sizes[0] / FEAT;   // 100000
  const int E = in_sizes[1];          // 1250000

  // workspace layout (all 4-byte elements, ~7.0 MB total)
  int*   in_cnt   = (int*)d_ws;            // N
  int*   out_cnt  = in_cnt + N;            // N
  int*   rowptr   = out_cnt + N;           // N+1
  int*   cursor   = rowptr + (N + 1);      // N
  int*   bsum     = cursor + N;            // scan block sums (<= 4096)
  int*   col      = bsum + 4096;           // E
  float* src_norm = (float*)(col + E);     // N

  const int T = 256;
  const int nb_scan = (N + SCAN_T * SCAN_V - 1) / (SCAN_T * SCAN_V);

  // p1: zero counters (fresh every call; atomics accumulate otherwise)
  k_zero_counts<<<(2 * N + T - 1) / T, T, 0, stream>>>(in_cnt, 2 * N);

  // p2: degree histograms
  k_count<<<(E + T - 1) / T, T, 0, stream>>>(src, dst, out_cnt, in_cnt, E);

  // p3: source norms
  k_srcnorm<<<(N + T - 1) / T, T, 0, stream>>>(out_cnt, src_norm, N);

  // p4: exclusive scan in_cnt -> rowptr (+ cursor copy)
  k_scan1<<<nb_scan, SCAN_T, 0, stream>>>(in_cnt, rowptr, bsum, N);
  k_scan_top<<<1, 32, 0, stream>>>(bsum, nb_scan);
  k_scan2<<<(N + T - 1) / T, T, 0, stream>>>(rowptr, bsum, cursor, N, E);

  // p5: group source indices by destination
  k_fill<<<(E + T - 1) / T, T, 0, stream>>>(src, dst, cursor, col, E);

  // p6: one wave per node, 12500 blocks x 8 waves = 100000 waves
  k_gather<<<(N + 7) / 8, T, 0, stream>>>(emb, col, rowptr, src_norm, out, N);
}